// Eikonal2D_15307263443052
// MI455X (gfx1250) — compile-verified
//
#include <hip/hip_runtime.h>
#include <hip/hip_bf16.h>

// ---------------- problem constants (match reference) ----------------
#define GN     512
#define BIGF   1.0e10f
#define SRC_I  256
#define SRC_J  256
#define NITER  1100

// ---------------- time-skew fusion ----------------
#define STEPS  10                   // fused Jacobi iterations per launch (even)
#define NOUTER (NITER / STEPS)      // 110 launches (10*110 == 1100 exactly)
#define PAD    STEPS                // BIG border width of padded u buffers
#define GP     (GN + 2 * PAD)       // 532 padded pitch

// ---------------- tiling ----------------
#define TX   64                     // output tile cols (j, contiguous)
#define TY   32                     // output tile rows (i)
#define BDX  64
#define BDY  8
#define NTHR (BDX * BDY)            // 512 threads = 16 waves (wave32)
#define LR   (TY + 2 * STEPS)       // 52 LDS rows
#define LC   (TX + 2 * STEPS)       // 84 LDS cols
#define LCP  (LC + 1)               // 85: odd pitch, conflict-free
#define NL   (LR * LC)              // 4368 halo elements

// ---------------- CDNA5 async global->LDS path (probe-verified) ----------------
#if defined(__has_builtin)
#if __has_builtin(__builtin_amdgcn_global_load_async_to_lds_b32)
#define USE_ASYNC_LDS 1
#endif
#if __has_builtin(__builtin_amdgcn_s_wait_asynccnt)
#define HAVE_WAIT_ASYNC 1
#endif
#endif

typedef __attribute__((address_space(1))) int* gptr_i32;
typedef __attribute__((address_space(3))) int* lptr_i32;

__device__ __forceinline__ void async_wait_all() {
#if defined(USE_ASYNC_LDS)
#if defined(HAVE_WAIT_ASYNC)
    __builtin_amdgcn_s_wait_asynccnt(0);
#else
    asm volatile("s_wait_asynccnt 0" ::: "memory");
#endif
#endif
}

__global__ void eik_init(float* __restrict__ a, float* __restrict__ b) {
    int idx = blockIdx.x * blockDim.x + threadIdx.x;
    if (idx < GP * GP) {
        a[idx] = (idx == (SRC_I + PAD) * GP + (SRC_J + PAD)) ? 0.0f : BIGF;
        b[idx] = BIGF;   // border ring stays BIG forever; interior overwritten each step
    }
}

// One launch == STEPS exact global Jacobi iterations (time-skewed, redundant halo).
__global__ __launch_bounds__(NTHR)
void eik_step(const float* __restrict__ ucur,
              float* __restrict__ unxt,
              const float* __restrict__ f) {
    __shared__ float buf[2][LR][LCP];   // ping-pong u tiles
    __shared__ float fbuf[LR][LCP];     // slowness tile (H == 1.0 so fh == f)

    const int bj  = blockIdx.x * TX;    // tile origin, grid coords
    const int bi  = blockIdx.y * TY;
    const int tid = threadIdx.y * BDX + threadIdx.x;

    // halo origin in padded coords: (bi - STEPS + PAD, bj - STEPS + PAD) == (bi, bj)
    const float* __restrict__ ubase = ucur + bi * GP + bj;

    for (int k = tid; k < NL; k += NTHR) {
        const int r = k / LC;
        const int c = k - r * LC;
#if defined(USE_ASYNC_LDS)
        __builtin_amdgcn_global_load_async_to_lds_b32(
            (gptr_i32)(ubase + r * GP + c), (lptr_i32)&buf[0][r][c], 0, 0);
#else
        buf[0][r][c] = ubase[r * GP + c];
#endif
        // slowness tile: only in-grid cells are ever read; mask the rest off
        const int gi = bi + r - STEPS;
        const int gj = bj + c - STEPS;
        if ((unsigned)gi < GN && (unsigned)gj < GN) {
#if defined(USE_ASYNC_LDS)
            __builtin_amdgcn_global_load_async_to_lds_b32(
                (gptr_i32)(f + gi * GN + gj), (lptr_i32)&fbuf[r][c], 0, 0);
#else
            fbuf[r][c] = f[gi * GN + gj];
#endif
        }
    }
    async_wait_all();
    __syncthreads();

    // Inner Jacobi steps: region shrinks by one ring per step.
#pragma unroll
    for (int s = 0; s < STEPS; ++s) {
        const int lo   = 1 + s;
        const int rows = LR - 2 * lo;       // 50,48,...,32
        const int cols = LC - 2 * lo;       // 82,80,...,64
        const int cnt  = rows * cols;
        const int p    = s & 1;
        for (int k = tid; k < cnt; k += NTHR) {
            const int r = lo + k / cols;    // cols is compile-time const per unrolled s
            const int c = lo + k - (k / cols) * cols;
            const int gi = bi + r - STEPS;
            const int gj = bj + c - STEPS;
            const float uc = buf[p][r][c];
            float res = uc;                 // out-of-grid cells stay BIG
            if ((unsigned)gi < GN && (unsigned)gj < GN) {
                const float a  = fminf(buf[p][r - 1][c], buf[p][r + 1][c]); // i-dir
                const float b  = fminf(buf[p][r][c - 1], buf[p][r][c + 1]); // j-dir
                const float fh = fbuf[r][c];
                const float d  = fabsf(a - b);
                const float u1 = fminf(a, b) + fh;
                const float sq = fmaxf(2.0f * fh * fh - d * d, 0.0f);
                const float u2 = 0.5f * (a + b + sqrtf(sq));
                const float un = (d >= fh) ? u1 : u2;
                res = fminf(uc, un);
                if (gi == SRC_I && gj == SRC_J) res = 0.0f;
            }
            buf[p ^ 1][r][c] = res;
        }
        __syncthreads();
    }

    // STEPS is even -> final iterate lives in buf[0]; write the TXxTY tile.
    const int lj = threadIdx.x;
#pragma unroll
    for (int q = 0; q < TY / BDY; ++q) {
        const int li = threadIdx.y + q * BDY;
        const int gi = bi + li;
        const int gj = bj + lj;
        unxt[(gi + PAD) * GP + (gj + PAD)] = buf[0][STEPS + li][STEPS + lj];
    }
}

__global__ void eik_extract(const float* __restrict__ u, float* __restrict__ out) {
    int idx = blockIdx.x * blockDim.x + threadIdx.x;
    if (idx < GN * GN) {
        const int i = idx / GN;
        const int j = idx - i * GN;
        out[idx] = u[(i + PAD) * GP + (j + PAD)];
    }
}

extern "C" void kernel_launch(void* const* d_in, const int* in_sizes, int n_in,
                              void* d_out, int out_size, void* d_ws, size_t ws_size,
                              hipStream_t stream) {
    const float* f = (const float*)d_in[0];

    float* A = (float*)d_ws;
    float* B = A + (size_t)GP * GP;     // 283024 floats each (~2.26 MB total)

    eik_init<<<(GP * GP + 255) / 256, 256, 0, stream>>>(A, B);

    dim3 grid(GN / TX, GN / TY);        // 8 x 16 = 128 blocks
    dim3 blk(BDX, BDY);

    float* cur = A;
    float* nxt = B;
    for (int it = 0; it < NOUTER; ++it) {
        eik_step<<<grid, blk, 0, stream>>>(cur, nxt, f);
        float* t = cur; cur = nxt; nxt = t;
    }
    eik_extract<<<(GN * GN + 255) / 256, 256, 0, stream>>>(cur, (float*)d_out);
}